// NNUENetworkSparse_21835613733381
// MI455X (gfx1250) — compile-verified
//
#include <hip/hip_runtime.h>
#include <hip/hip_bf16.h>

typedef __attribute__((ext_vector_type(2))) float v2f;
typedef __attribute__((ext_vector_type(8))) float v8f;

#define BATCH   16384
#define FEATS   32
#define TILE    16          // samples (rows) per block = one WMMA M-tile
#define HSTRIDE 516         // padded LDS row stride for hidden tile (floats); 516%64==4 -> conflict-free frags
#define XSTRIDE 34          // padded LDS row stride for 16x32 activation tile

__device__ __forceinline__ float clip01(float v) {
    return fminf(fmaxf(v, 0.0f), 1.0f);
}

__global__ __launch_bounds__(256) void nnue_fused_kernel(
    const int*   __restrict__ wIdx,  // (BATCH*32)
    const int*   __restrict__ bIdx,  // (BATCH*32)
    const float* __restrict__ stm,   // (BATCH,1)
    const float* __restrict__ ftW,   // (40960,256)
    const float* __restrict__ ftB,   // (256)
    const float* __restrict__ l1W,   // (32,512)
    const float* __restrict__ l1B,   // (32)
    const float* __restrict__ l2W,   // (32,32)
    const float* __restrict__ l2B,   // (32)
    const float* __restrict__ l3W,   // (1,32)
    const float* __restrict__ l3B,   // (1)
    float*       __restrict__ out)   // (BATCH,1)
{
    __shared__ __align__(16) float hid[TILE * HSTRIDE];   // 16 x 512 hidden tile (padded)
    __shared__ int   sIdx[TILE * 64];                     // [sample][bag][32]
    __shared__ float sStm[TILE];
    __shared__ __align__(8) float xbuf[16 * XSTRIDE];     // 16x32 activation tile (padded)

    const int tid        = threadIdx.x;      // 0..255
    const int sampleBase = blockIdx.x * TILE;

    // ---- Phase 0: stage indices + stm into LDS --------------------------------
    #pragma unroll
    for (int r = 0; r < 4; ++r) {
        int j      = r * 256 + tid;          // 0..1023
        int sample = j >> 6;                 // 64 idx per sample (32 white + 32 black)
        int slot   = j & 63;
        int bag    = slot >> 5;
        int i      = slot & 31;
        const int* src = bag ? bIdx : wIdx;
        sIdx[j] = src[(sampleBase + sample) * FEATS + i];
    }
    if (tid < TILE) sStm[tid] = stm[sampleBase + tid];
    __syncthreads();

    // ---- Phase 1: EmbeddingBag sums (dominant cost, L2-bound gather) ----------
    // 16 samples x 2 bags x 256 dims = 2048 float4 chunks; 8 per thread.
    #pragma unroll
    for (int q = 0; q < 8; ++q) {
        int chunk  = q * 256 + tid;          // 0..2047
        int sample = chunk >> 7;             // 128 chunks per sample
        int c      = chunk & 127;
        int bag    = c >> 6;                 // 0 = white, 1 = black
        int dim    = (c & 63) * 4;           // 0..252
        const int* idxp = &sIdx[sample * 64 + bag * 32];

        float ax = 0.f, ay = 0.f, az = 0.f, aw = 0.f;
        #pragma unroll 4
        for (int i = 0; i < 32; ++i) {
            size_t row = (size_t)idxp[i];
            const float4 v = *(const float4*)(ftW + (row << 8) + dim);  // global_load_b128
            ax += v.x; ay += v.y; az += v.z; aw += v.w;
        }
        const float4 bb = *(const float4*)(ftB + dim);
        float4 acc;
        acc.x = clip01(ax + bb.x);
        acc.y = clip01(ay + bb.y);
        acc.z = clip01(az + bb.z);
        acc.w = clip01(aw + bb.w);

        // stm concat swap: stm=1 -> [white|black], stm=0 -> [black|white]
        bool stmb = sStm[sample] > 0.5f;
        int  off  = ((bag ^ (stmb ? 0 : 1)) ? 256 : 0);
        *(float4*)(&hid[sample * HSTRIDE + off + dim]) = acc;
    }
    __syncthreads();

    // ---- Phase 2: MLP tail on wave 0 only (negligible work) -------------------
    if (tid >= 32) return;

    const int lane = tid;          // wave32 lane
    const int lm   = lane & 15;    // M (A) / N (B,D) index
    const int kh   = lane >> 4;    // K-half selector (K pair 0-1 vs 2-3)

    // ---- Layer 1: x = clip(hidden(16x512) @ l1_w.T + l1_b), two 16x16 N-tiles
    v8f c0, c1;
    {
        float b0 = l1B[lm], b1 = l1B[16 + lm];
        #pragma unroll
        for (int j = 0; j < 8; ++j) { c0[j] = b0; c1[j] = b1; }
    }
    const float* arow = &hid[lm * HSTRIDE + 2 * kh];
    const float* w0   = l1W + (size_t)lm * 512 + 2 * kh;
    const float* w1   = l1W + (size_t)(16 + lm) * 512 + 2 * kh;
    #pragma unroll 8
    for (int k = 0; k < 512; k += 4) {
        v2f a;   a.x   = arow[k]; a.y   = arow[k + 1];   // ds_load_b64, conflict-free
        v2f bf0; bf0.x = w0[k];   bf0.y = w0[k + 1];
        v2f bf1; bf1.x = w1[k];   bf1.y = w1[k + 1];
        c0 = __builtin_amdgcn_wmma_f32_16x16x4_f32(false, a, false, bf0, (short)0, c0, false, false);
        c1 = __builtin_amdgcn_wmma_f32_16x16x4_f32(false, a, false, bf1, (short)0, c1, false, false);
    }
    // clip + spill x (16x32) to LDS to re-form A fragments for layer 2
    #pragma unroll
    for (int j = 0; j < 8; ++j) {
        int m = j + 8 * kh;                               // D layout: M = j + 8*(lane/16)
        xbuf[m * XSTRIDE + lm]      = clip01(c0[j]);
        xbuf[m * XSTRIDE + 16 + lm] = clip01(c1[j]);
    }

    // ---- Layer 2: y = clip(x(16x32) @ l2_w.T + l2_b)
    v8f d0, d1;
    {
        float b0 = l2B[lm], b1 = l2B[16 + lm];
        #pragma unroll
        for (int j = 0; j < 8; ++j) { d0[j] = b0; d1[j] = b1; }
    }
    const float* xrow = &xbuf[lm * XSTRIDE + 2 * kh];
    const float* u0   = l2W + lm * 32 + 2 * kh;
    const float* u1   = l2W + (16 + lm) * 32 + 2 * kh;
    #pragma unroll
    for (int k = 0; k < 32; k += 4) {
        v2f a;   a.x   = xrow[k]; a.y   = xrow[k + 1];
        v2f bf0; bf0.x = u0[k];   bf0.y = u0[k + 1];
        v2f bf1; bf1.x = u1[k];   bf1.y = u1[k + 1];
        d0 = __builtin_amdgcn_wmma_f32_16x16x4_f32(false, a, false, bf0, (short)0, d0, false, false);
        d1 = __builtin_amdgcn_wmma_f32_16x16x4_f32(false, a, false, bf1, (short)0, d1, false, false);
    }
    // clip + spill y (reuse xbuf; all reads of x completed above, DS ops in-order per wave)
    #pragma unroll
    for (int j = 0; j < 8; ++j) {
        int m = j + 8 * kh;
        xbuf[m * XSTRIDE + lm]      = clip01(d0[j]);
        xbuf[m * XSTRIDE + 16 + lm] = clip01(d1[j]);
    }

    // ---- Layer 3: out = y @ l3_w.T + l3_b  (per-row 32-wide dot, lanes 0..15)
    if (lane < 16) {
        float s = l3B[0];
        #pragma unroll
        for (int n = 0; n < 32; ++n)
            s += xbuf[lane * XSTRIDE + n] * l3W[n];
        out[sampleBase + lane] = s;
    }
}

extern "C" void kernel_launch(void* const* d_in, const int* in_sizes, int n_in,
                              void* d_out, int out_size, void* d_ws, size_t ws_size,
                              hipStream_t stream) {
    (void)in_sizes; (void)n_in; (void)out_size; (void)d_ws; (void)ws_size;
    // setup_inputs order:
    // 0 white_indices, 1 white_offsets, 2 black_indices, 3 black_offsets, 4 stm,
    // 5 ft_w, 6 ft_b, 7 l1_w, 8 l1_b, 9 l2_w, 10 l2_b, 11 l3_w, 12 l3_b
    const int*   wIdx = (const int*)d_in[0];
    const int*   bIdx = (const int*)d_in[2];   // offsets are a fixed stride of 32; unused
    const float* stm  = (const float*)d_in[4];
    const float* ftW  = (const float*)d_in[5];
    const float* ftB  = (const float*)d_in[6];
    const float* l1W  = (const float*)d_in[7];
    const float* l1B  = (const float*)d_in[8];
    const float* l2W  = (const float*)d_in[9];
    const float* l2B  = (const float*)d_in[10];
    const float* l3W  = (const float*)d_in[11];
    const float* l3B  = (const float*)d_in[12];
    float* out = (float*)d_out;

    dim3 grid(BATCH / TILE);   // 1024 blocks
    dim3 block(256);           // 8 waves; wave 0 runs the WMMA MLP tail
    nnue_fused_kernel<<<grid, block, 0, stream>>>(
        wIdx, bIdx, stm, ftW, ftB, l1W, l1B, l2W, l2B, l3W, l3B, out);
}